// ConnectFourFeatures_3582002725045
// MI455X (gfx1250) — compile-verified
//
#include <hip/hip_runtime.h>
#include <stdint.h>

// ---------------------------------------------------------------------------
// ConnectFourFeatures: out = concat([x, 8 per-channel run-sum stencils], C)
// x: (131072, 3, 6, 7) f32 -> out: (131072, 27, 6, 7) f32
// ~658 MB total traffic -> ~28us @ 23.3 TB/s: pure bandwidth problem.
// Design: TDM streams tiles HBM<->LDS (1-D load, 2-D strided stores);
// every thread runs identical straight-line code (no wave divergence):
// one (img,ch) plane in registers -> all 9 output planes.
// ---------------------------------------------------------------------------

constexpr int CIN  = 3;
constexpr int H    = 6;
constexpr int W    = 7;
constexpr int HW   = H * W;             // 42
constexpr int IN_PER  = CIN * HW;       // 126 dwords per image in
constexpr int COUT    = 27;
constexpr int OUT_PER = COUT * HW;      // 1134 dwords per image out
constexpr int TILE    = 32;             // images per block (divides 131072)
constexpr int THREADS = TILE * CIN;     // 96 threads = 3 waves, one plane each
constexpr int BUF_DW  = TILE * IN_PER;  // 4032 dwords = 16128 B per buffer
constexpr int NGROUP  = 9;              // passthrough + 8 features

typedef unsigned int v4u __attribute__((ext_vector_type(4)));
typedef int          v4i __attribute__((ext_vector_type(4)));
typedef int          v8i __attribute__((ext_vector_type(8)));

#if __has_builtin(__builtin_amdgcn_tensor_load_to_lds) && \
    __has_builtin(__builtin_amdgcn_tensor_store_from_lds) && \
    __has_builtin(__builtin_amdgcn_s_wait_tensorcnt)
#define USE_TDM 1
#else
#define USE_TDM 0
#endif

#if USE_TDM
// 2-D TDM descriptor (D#): tile0 dwords per row, tile1 rows, row stride
// stride0 dwords. Bitfields per CDNA5 ISA.
__device__ __forceinline__ void tdm_build(v4u& g0, v8i& g1, uint32_t lds_off,
                                          uint64_t gaddr, uint32_t tile0,
                                          uint32_t tile1, uint32_t td0,
                                          uint32_t td1, uint32_t stride0) {
  g0.x = 1u;                                        // count=1, user mode
  g0.y = lds_off;                                   // lds_addr
  g0.z = (uint32_t)gaddr;                           // global_addr lo
  g0.w = (uint32_t)((gaddr >> 32) & 0x1FFFFFFu)     // global_addr[56:32]
         | (2u << 30);                              // type=2 ("image")
  g1[0] = (int)(2u << 16);                          // data_size=4B, mask=0
  g1[1] = (int)((td0 & 0xFFFFu) << 16);             // tensor_dim0[15:0]
  g1[2] = (int)((td0 >> 16) | ((td1 & 0xFFFFu) << 16));  // td0 hi | td1 lo
  g1[3] = (int)((td1 >> 16) | (tile0 << 16));       // td1 hi | tile_dim0
  g1[4] = (int)(tile1 & 0xFFFFu);                   // tile_dim1 | tile_dim2=0
  g1[5] = (int)stride0;                             // tensor_dim0_stride lo32
  g1[6] = 0;                                        // stride hi, dim1_stride
  g1[7] = 0;
}

__device__ __forceinline__ void tdm_load(const v4u& g0, const v8i& g1) {
  v4i z4 = {0, 0, 0, 0};
#if defined(__clang_major__) && __clang_major__ >= 23
  v8i z8 = {0, 0, 0, 0, 0, 0, 0, 0};
  __builtin_amdgcn_tensor_load_to_lds(g0, g1, z4, z4, z8, 0);
#else
  __builtin_amdgcn_tensor_load_to_lds(g0, g1, z4, z4, 0);
#endif
}

__device__ __forceinline__ void tdm_store(const v4u& g0, const v8i& g1) {
  v4i z4 = {0, 0, 0, 0};
#if defined(__clang_major__) && __clang_major__ >= 23
  v8i z8 = {0, 0, 0, 0, 0, 0, 0, 0};
  __builtin_amdgcn_tensor_store_from_lds(g0, g1, z4, z4, z8, 0);
#else
  __builtin_amdgcn_tensor_store_from_lds(g0, g1, z4, z4, 0);
#endif
}
#endif  // USE_TDM

// Run-sum along (DH,DW) with radius P, zero padding; compile-time bounds.
template <int DH, int DW, int P>
__device__ __forceinline__ void stencil(const float v[HW], float o[HW]) {
#pragma unroll
  for (int h = 0; h < H; ++h) {
#pragma unroll
    for (int w = 0; w < W; ++w) {
      float s = 0.0f;
#pragma unroll
      for (int d = -P; d <= P; ++d) {
        const int hh = h + d * DH;
        const int ww = w + d * DW;
        if (hh >= 0 && hh < H && ww >= 0 && ww < W) s += v[hh * W + ww];
      }
      o[h * W + w] = s;
    }
  }
}

template <int G>
__device__ __forceinline__ void compute_group(const float v[HW], float o[HW]) {
  if constexpr (G == 0) {
#pragma unroll
    for (int k = 0; k < HW; ++k) o[k] = v[k];    // passthrough channels
  } else if constexpr (G == 1) stencil<0, 1, 1>(v, o);   // horiz l=3
  else if constexpr (G == 2) stencil<1, 0, 1>(v, o);     // vert  l=3
  else if constexpr (G == 3) stencil<1, 1, 1>(v, o);     // diag  l=3
  else if constexpr (G == 4) stencil<1, -1, 1>(v, o);    // anti  l=3
  else if constexpr (G == 5) stencil<0, 1, 2>(v, o);     // horiz l=5
  else if constexpr (G == 6) stencil<1, 0, 2>(v, o);     // vert  l=5
  else if constexpr (G == 7) stencil<1, 1, 2>(v, o);     // diag  l=5
  else stencil<1, -1, 2>(v, o);                          // anti  l=5
}

__global__ void __launch_bounds__(THREADS)
connect4_features_kernel(const float* __restrict__ x, float* __restrict__ out) {
  const int tid = threadIdx.x;                       // == plane = img*3 + ch
  const long long img0 = (long long)blockIdx.x * TILE;

#if USE_TDM
  __shared__ float buf[2][BUF_DW];                   // 32.25 KB total

  // ---- Stage in: one contiguous 16KB tensor_load_to_lds into buf[0] -------
  if (tid < 32) {
    v4u g0; v8i g1;
    tdm_build(g0, g1, (uint32_t)(uintptr_t)&buf[0][0],
              (uint64_t)(uintptr_t)(x + img0 * IN_PER),
              BUF_DW, 1, BUF_DW, 1, BUF_DW);
    tdm_load(g0, g1);
    __builtin_amdgcn_s_wait_tensorcnt(0);
  }
  __syncthreads();
  asm volatile("" ::: "memory");   // LDS written behind the compiler's back

  // ---- Pull this thread's (img,ch) plane into registers -------------------
  float v[HW];
  {
    const float2* s = (const float2*)&buf[0][tid * HW];  // 168B-aligned
#pragma unroll
    for (int k = 0; k < HW / 2; ++k) {
      float2 t = s[k];
      v[2 * k] = t.x;
      v[2 * k + 1] = t.y;
    }
  }
  __syncthreads();   // everyone done reading buf[0] before group 0 reuses it

  // ---- 9 channel-triple groups, ping-pong staged, 2-D TDM stores ----------
  // Group g occupies out channels [3g, 3g+3) => dword offset g*126 per image;
  // rows of 126 dwords per image at stride 1134, 32 rows per store.
  const float* out_base = out + img0 * OUT_PER;
#define DO_GROUP(G)                                                           \
  {                                                                           \
    float o[HW];                                                              \
    compute_group<G>(v, o);                                                   \
    float2* d2 = (float2*)&buf[(G) & 1][tid * HW];                            \
    _Pragma("unroll") for (int k = 0; k < HW / 2; ++k) {                      \
      float2 r; r.x = o[2 * k]; r.y = o[2 * k + 1]; d2[k] = r;                \
    }                                                                         \
    asm volatile("" ::: "memory");                                            \
    __syncthreads();                                                          \
    if (tid < 32) {                                                           \
      v4u g0; v8i g1;                                                         \
      tdm_build(g0, g1, (uint32_t)(uintptr_t)&buf[(G) & 1][0],                \
                (uint64_t)(uintptr_t)(out_base + (G) * (CIN * HW)),           \
                IN_PER, TILE, OUT_PER, TILE, OUT_PER);                        \
      tdm_store(g0, g1);                                                      \
      __builtin_amdgcn_s_wait_tensorcnt(1); /* store(G-1) done -> buffer */   \
    }                                                                         \
    __syncthreads();                                                          \
  }
  DO_GROUP(0) DO_GROUP(1) DO_GROUP(2) DO_GROUP(3) DO_GROUP(4)
  DO_GROUP(5) DO_GROUP(6) DO_GROUP(7) DO_GROUP(8)
#undef DO_GROUP
  if (tid < 32) __builtin_amdgcn_s_wait_tensorcnt(0);  // before LDS teardown

#else  // ---- Fallback without TDM: direct global traffic -------------------
  float v[HW];
  {
    const float2* s = (const float2*)(x + (img0 * CIN + tid) * HW);
#pragma unroll
    for (int k = 0; k < HW / 2; ++k) {
      float2 t = s[k];
      v[2 * k] = t.x;
      v[2 * k + 1] = t.y;
    }
  }
  const int img = tid / CIN, ch = tid % CIN;
  float* ob = out + (img0 + img) * OUT_PER + ch * HW;
#define DO_GROUP(G)                                                           \
  {                                                                           \
    float o[HW];                                                              \
    compute_group<G>(v, o);                                                   \
    float2* d2 = (float2*)(ob + (G) * (CIN * HW));                            \
    _Pragma("unroll") for (int k = 0; k < HW / 2; ++k) {                      \
      float2 r; r.x = o[2 * k]; r.y = o[2 * k + 1]; d2[k] = r;                \
    }                                                                         \
  }
  DO_GROUP(0) DO_GROUP(1) DO_GROUP(2) DO_GROUP(3) DO_GROUP(4)
  DO_GROUP(5) DO_GROUP(6) DO_GROUP(7) DO_GROUP(8)
#undef DO_GROUP
#endif
}

extern "C" void kernel_launch(void* const* d_in, const int* in_sizes, int n_in,
                              void* d_out, int out_size, void* d_ws, size_t ws_size,
                              hipStream_t stream) {
  const float* x = (const float*)d_in[0];  // filters d_in[1..8] are fixed 0/1
                                           // patterns, folded into the kernel
  float* out = (float*)d_out;
  const int n_img = in_sizes[0] / IN_PER;  // 131072
  const int blocks = n_img / TILE;         // 4096
  hipLaunchKernelGGL(connect4_features_kernel, dim3(blocks), dim3(THREADS), 0,
                     stream, x, out);
}